// BiLIFNode_33904471834955
// MI455X (gfx1250) — compile-verified
//
#include <hip/hip_runtime.h>

// BiLIF forward: out[t] = (heaviside(h_fwd[t] - 0.75) + heaviside(h_rev[t] - 1.25)) / 2
// h = v + (x - v) * 0.75 ;  hard reset v = s ? 0 : h
// x: [T=16, B*N = 4194304] fp32.  Pure streaming: 256MB in + 256MB out (~22us @ 23.3TB/s).
// Data path: gfx1250 async global->LDS staging (ASYNCcnt) + NT-hinted b128 stores.

#define TSTEPS 16
#define BLOCK  256

#if defined(__gfx1250__) && __has_builtin(__builtin_amdgcn_global_load_async_to_lds_b128)
#define USE_ASYNC_LDS 1
#else
#define USE_ASYNC_LDS 0
#endif

typedef int   v4i __attribute__((ext_vector_type(4)));
typedef float v4f __attribute__((ext_vector_type(4)));
typedef __attribute__((address_space(1))) v4i gv4i;   // global (AS1)
typedef __attribute__((address_space(3))) v4i lv4i;   // LDS (AS3)

__device__ __forceinline__ float lif2_step(float xf, float xr, float& v1, float& v2) {
    const float k = 0.75f;            // 1/tau
    // forward neuron, v_th = 0.75
    float h1 = __builtin_fmaf(xf - v1, k, v1);
    bool  f1 = (h1 >= 0.75f);
    float s1 = f1 ? 1.0f : 0.0f;
    v1       = f1 ? 0.0f : h1;        // hard reset to 0
    // reverse-stream neuron, v_th = 1.25
    float h2 = __builtin_fmaf(xr - v2, k, v2);
    bool  f2 = (h2 >= 1.25f);
    float s2 = f2 ? 1.0f : 0.0f;
    v2       = f2 ? 0.0f : h2;
    return (s1 + s2) * 0.5f;
}

__global__ __launch_bounds__(BLOCK) void bilif_kernel(const float* __restrict__ x,
                                                      float* __restrict__ out,
                                                      int stride4 /* B*N/4 */) {
    const int tid  = threadIdx.x;
    const int col4 = blockIdx.x * BLOCK + tid;     // float4 column index
    if (col4 >= stride4) return;

    const v4f* __restrict__ xg = (const v4f*)x;
    v4f* __restrict__       og = (v4f*)out;

#if USE_ASYNC_LDS
    // Stage the whole 16-timestep tile for this block into LDS with the
    // gfx1250 async global->LDS path (tracked by ASYNCcnt, no VGPR staging).
    // cpol=1 -> TH=NT: inbound stream is read exactly once, keep it out of L2.
    __shared__ v4f tile[TSTEPS * BLOCK];           // 64 KB -> 5 blocks/WGP, 40 waves
#pragma unroll
    for (int t = 0; t < TSTEPS; ++t) {
        __builtin_amdgcn_global_load_async_to_lds_b128(
            (gv4i*)(xg + (size_t)t * stride4 + col4),
            (lv4i*)&tile[t * BLOCK + tid],
            /*imm offset*/ 0, /*cpol: TH_NT*/ 1);
    }
#if __has_builtin(__builtin_amdgcn_s_wait_asynccnt)
    __builtin_amdgcn_s_wait_asynccnt(0);
#else
    asm volatile("s_wait_asynccnt 0" ::: "memory");
#endif
#else
    // Fallback: stage all timesteps in registers (64 VGPRs of input).
    v4f xin[TSTEPS];
#pragma unroll
    for (int t = 0; t < TSTEPS; ++t)
        xin[t] = __builtin_nontemporal_load(xg + (size_t)t * stride4 + col4);
#endif

    float v1[4] = {0.f, 0.f, 0.f, 0.f};
    float v2[4] = {0.f, 0.f, 0.f, 0.f};

#pragma unroll
    for (int t = 0; t < TSTEPS; ++t) {
#if USE_ASYNC_LDS
        v4f xf = tile[t * BLOCK + tid];                 // forward input  x[t]
        v4f xr = tile[(TSTEPS - 1 - t) * BLOCK + tid];  // reverse input  x[T-1-t]
#else
        v4f xf = xin[t];
        v4f xr = xin[TSTEPS - 1 - t];
#endif
        v4f o;
#pragma unroll
        for (int j = 0; j < 4; ++j)
            o[j] = lif2_step(xf[j], xr[j], v1[j], v2[j]);
        // Output is write-once, never re-read: non-temporal b128 store.
        __builtin_nontemporal_store(o, og + (size_t)t * stride4 + col4);
    }
}

extern "C" void kernel_launch(void* const* d_in, const int* in_sizes, int n_in,
                              void* d_out, int out_size, void* d_ws, size_t ws_size,
                              hipStream_t stream) {
    (void)n_in; (void)out_size; (void)d_ws; (void)ws_size;
    const float* x   = (const float*)d_in[0];
    float*       out = (float*)d_out;

    const int total   = in_sizes[0];        // T * B * N = 67108864
    const int stride  = total / TSTEPS;     // B * N     = 4194304
    const int stride4 = stride / 4;         // 1048576 float4 columns

    const int nblocks = (stride4 + BLOCK - 1) / BLOCK;  // 4096
    bilif_kernel<<<nblocks, BLOCK, 0, stream>>>(x, out, stride4);
}